// DoubleNet_1090921693356
// MI455X (gfx1250) — compile-verified
//
#include <hip/hip_runtime.h>
#include <math.h>

// ---------------------------------------------------------------------------
// Constants fixed by the reference model
// ---------------------------------------------------------------------------
#define HH   3        // heads
#define DD   128      // embedding per head
#define CC   (HH*DD)  // 384

typedef float v2f __attribute__((ext_vector_type(2)));
typedef float v8f __attribute__((ext_vector_type(8)));

__device__ __forceinline__ v8f wmma4(v2f a, v2f b, v8f c) {
  return __builtin_amdgcn_wmma_f32_16x16x4_f32(
      /*neg_a=*/false, a, /*neg_b=*/false, b,
      /*c_mod=*/(short)0, c, /*reuse_a=*/false, /*reuse_b=*/false);
}

// ---------------------------------------------------------------------------
// WMMA fp32 GEMM: C[M,Nc] = A[M,K] @ B[K,Nc] (+ bias[Nc] if non-null)
// One wave32 per 16x64 output strip (4 adjacent 16x16 N-tiles):
//   - the A fragment is reused across 4 WMMAs (4x fewer A loads / FLOP)
//   - 4 independent accumulator chains give the XDL pipe back-to-back work
// K stepped by 4 via V_WMMA_F32_16X16X4_F32 (full fp32 -> matches reference
// precision; bf16 WMMA would be faster but GEMMs are not the bottleneck here).
// Layouts per CDNA5 ISA 7.12.2:
//   A 16x4 : lanes 0-15 -> M=lane, v0=K0 v1=K1 ; lanes 16-31 -> M=lane-16, v0=K2 v1=K3
//   B 4x16 : lanes 0-15 -> N=lane, v0=K0 v1=K1 ; lanes 16-31 -> N=lane-16, v0=K2 v1=K3
//   C 16x16: VGPR v -> row = v + 8*(lane>=16), col = lane&15
// ---------------------------------------------------------------------------
__global__ void wmma_gemm_f32(const float* __restrict__ A,
                              const float* __restrict__ B,
                              const float* __restrict__ bias,
                              float* __restrict__ C,
                              int M, int K, int Nc) {
  int wave   = (int)((blockIdx.x * blockDim.x + threadIdx.x) >> 5);
  int lane   = threadIdx.x & 31;
  int nstrip = Nc >> 6;                    // Nc / 64 (384 and 128 both divide)
  int tm     = wave / nstrip;
  int ts     = wave - tm * nstrip;
  if (tm * 16 >= M) return;                // wave-uniform: EXEC stays all-1s

  const int m0 = tm << 4, n0 = ts << 6;
  const int hl = lane >> 4;                // 0: K pair (0,1), 1: K pair (2,3)
  const int l  = lane & 15;

  v8f acc0 = {}, acc1 = {}, acc2 = {}, acc3 = {};
  const float* arow = A + (size_t)(m0 + l) * K + 2 * hl;
  const float* brow = B + (size_t)(2 * hl) * Nc + n0 + l;
  for (int k = 0; k < K; k += 4) {
    v2f a;
    a.x = arow[0];
    a.y = arow[1];
    v2f b0, b1, b2, b3;
    b0.x = brow[0];  b0.y = brow[Nc + 0];
    b1.x = brow[16]; b1.y = brow[Nc + 16];
    b2.x = brow[32]; b2.y = brow[Nc + 32];
    b3.x = brow[48]; b3.y = brow[Nc + 48];
    acc0 = wmma4(a, b0, acc0);
    acc1 = wmma4(a, b1, acc1);
    acc2 = wmma4(a, b2, acc2);
    acc3 = wmma4(a, b3, acc3);
    arow += 4;
    brow += (size_t)4 * Nc;
  }

  for (int v = 0; v < 8; ++v) {
    int row = m0 + v + 8 * hl;
    float* crow = C + (size_t)row * Nc + n0 + l;
    float b0v = 0.f, b1v = 0.f, b2v = 0.f, b3v = 0.f;
    if (bias) {
      const float* bp = bias + n0 + l;
      b0v = bp[0]; b1v = bp[16]; b2v = bp[32]; b3v = bp[48];
    }
    crow[0]  = acc0[v] + b0v;
    crow[16] = acc1[v] + b1v;
    crow[32] = acc2[v] + b2v;
    crow[48] = acc3[v] + b3v;
  }
}

// ---------------------------------------------------------------------------
// el[n,h] = sum_d z[n,h,d]*al[h,d]; er likewise. One wave per (node, head).
// ---------------------------------------------------------------------------
__global__ void attn_lr(const float* __restrict__ z,
                        const float* __restrict__ al,
                        const float* __restrict__ ar,
                        float* __restrict__ el, float* __restrict__ er, int Nn) {
  int wave = (int)((blockIdx.x * blockDim.x + threadIdx.x) >> 5);
  int lane = threadIdx.x & 31;
  int n = wave / HH, h = wave - n * HH;
  if (n >= Nn) return;
  const float* zp  = z  + (size_t)n * CC + h * DD;
  const float* alp = al + h * DD;
  const float* arp = ar + h * DD;
  float sl = 0.f, sr = 0.f;
  for (int i = lane; i < DD; i += 32) {
    float zv = zp[i];
    sl = fmaf(zv, alp[i], sl);
    sr = fmaf(zv, arp[i], sr);
  }
  for (int o = 16; o; o >>= 1) {
    sl += __shfl_down(sl, o, 32);
    sr += __shfl_down(sr, o, 32);
  }
  if (lane == 0) { el[n * HH + h] = sl; er[n * HH + h] = sr; }
}

// ---------------------------------------------------------------------------
// Init per-layer buffers: m=-inf, den=0, xagg[n][c] = bias[c]
// ---------------------------------------------------------------------------
__global__ void init_agg(float* __restrict__ m, float* __restrict__ den,
                         float* __restrict__ xagg, const float* __restrict__ bias,
                         int Nn) {
  int t = blockIdx.x * blockDim.x + threadIdx.x;
  if (t < Nn * HH) { m[t] = -INFINITY; den[t] = 0.f; }
  if (t < Nn * CC) xagg[t] = bias[t % CC];
}

__global__ void zero_buf(float* __restrict__ p, int n) {
  int t = blockIdx.x * blockDim.x + threadIdx.x;
  if (t < n) p[t] = 0.f;
}

// Sign-aware float atomic max via int/uint ordering trick (init must be -inf).
__device__ inline void atomicMaxF(float* addr, float v) {
  if (v >= 0.f) atomicMax((int*)addr, __float_as_int(v));
  else          atomicMin((unsigned int*)addr, __float_as_uint(v));
}

// ---------------------------------------------------------------------------
// Edge pass 1: e = leaky_relu(el[src]+er[dst]); m[dst] = max(...)
// ---------------------------------------------------------------------------
__global__ void edge_max(const int* __restrict__ src, const int* __restrict__ dst,
                         const float* __restrict__ el, const float* __restrict__ er,
                         float* __restrict__ e_out, float* __restrict__ m, int Et) {
  int t = blockIdx.x * blockDim.x + threadIdx.x;
  if (t >= Et) return;
  int s = src[t], d = dst[t];
  for (int h = 0; h < HH; ++h) {
    float v = el[s * HH + h] + er[d * HH + h];
    v = (v > 0.f) ? v : 0.2f * v;           // NEG_SLOPE = 0.2
    e_out[t * HH + h] = v;
    atomicMaxF(&m[d * HH + h], v);
  }
}

// ---------------------------------------------------------------------------
// Edge pass 2: a = exp(e - m[dst]); den[dst] += a
// ---------------------------------------------------------------------------
__global__ void edge_exp(const int* __restrict__ dst,
                         const float* __restrict__ e, const float* __restrict__ m,
                         float* __restrict__ a, float* __restrict__ den, int Et) {
  int t = blockIdx.x * blockDim.x + threadIdx.x;
  if (t >= Et) return;
  int d = dst[t];
  for (int h = 0; h < HH; ++h) {
    float v = __expf(e[t * HH + h] - m[d * HH + h]);
    a[t * HH + h] = v;
    atomicAdd(&den[d * HH + h], v);
  }
}

// ---------------------------------------------------------------------------
// Edge pass 3: xagg[dst,h,:] += (a/den[dst,h]) * z[src,h,:]  (wave per edge)
// ---------------------------------------------------------------------------
__global__ void edge_aggr(const int* __restrict__ src, const int* __restrict__ dst,
                          const float* __restrict__ a, const float* __restrict__ den,
                          const float* __restrict__ z, float* __restrict__ xagg, int Et) {
  int wave = (int)((blockIdx.x * blockDim.x + threadIdx.x) >> 5);
  int lane = threadIdx.x & 31;
  if (wave >= Et) return;
  int s = src[wave], d = dst[wave];
  for (int h = 0; h < HH; ++h) {
    float w = a[wave * HH + h] / den[d * HH + h];
    const float4* zp = (const float4*)(z + (size_t)s * CC + h * DD);
    float* op = xagg + (size_t)d * CC + h * DD + lane * 4;
    float4 zv = zp[lane];
    atomicAdd(op + 0, w * zv.x);
    atomicAdd(op + 1, w * zv.y);
    atomicAdd(op + 2, w * zv.z);
    atomicAdd(op + 3, w * zv.w);
  }
}

// ---------------------------------------------------------------------------
// Pooling: pool[gid[n], c] += x[n, c]; cnt[gid[n]] += 1
// ---------------------------------------------------------------------------
__global__ void pool_sum(const float* __restrict__ x, const int* __restrict__ gid,
                         float* __restrict__ pool, float* __restrict__ cnt, int Nn) {
  int t = blockIdx.x * blockDim.x + threadIdx.x;
  if (t >= Nn * DD) return;
  int n = t >> 7, c = t & (DD - 1);
  int g = gid[n];
  atomicAdd(&pool[g * DD + c], x[(size_t)n * DD + c]);
  if (c == 0) atomicAdd(&cnt[g], 1.f);
}

// ---------------------------------------------------------------------------
// Final: out[g] = (poolA[g]/cntA[g]) . Wo[0:128] + (poolB[g]/cntB[g]) . Wo[128:256] + bo
// Wave per graph; write float64 (reference output dtype).
// ---------------------------------------------------------------------------
__global__ void final_out(const float* __restrict__ poolA, const float* __restrict__ cntA,
                          const float* __restrict__ poolB, const float* __restrict__ cntB,
                          const float* __restrict__ Wo, const float* __restrict__ bo,
                          double* __restrict__ out, int Gn) {
  int wave = (int)((blockIdx.x * blockDim.x + threadIdx.x) >> 5);
  int lane = threadIdx.x & 31;
  if (wave >= Gn) return;
  float ia = 1.f / cntA[wave], ib = 1.f / cntB[wave];
  float s = 0.f;
  for (int c = lane; c < DD; c += 32) {
    s = fmaf(poolA[wave * DD + c] * ia, Wo[c], s);
    s = fmaf(poolB[wave * DD + c] * ib, Wo[DD + c], s);
  }
  for (int o = 16; o; o >>= 1) s += __shfl_down(s, o, 32);
  if (lane == 0) out[wave] = (double)(s + bo[0]);
}

// ---------------------------------------------------------------------------
// Host-side orchestration
// ---------------------------------------------------------------------------
static inline void gemm(const float* A, const float* B, const float* bias, float* C,
                        int M, int K, int Nc, hipStream_t s) {
  int strips = (M / 16) * (Nc / 64);      // one wave per 16x64 strip
  int grid   = (strips + 7) / 8;          // 256 threads = 8 waves per block
  hipLaunchKernelGGL(wmma_gemm_f32, dim3(grid), dim3(256), 0, s, A, B, bias, C, M, K, Nc);
}

struct Scratch {
  float *z, *x, *y, *el, *er, *m, *den, *e, *a;
};

static void run_gat_layer(const float* xin, int Kin,
                          const int* src, const int* dst, int Et, int Nn,
                          const float* W, const float* al, const float* ar, const float* b,
                          const Scratch& ws, hipStream_t s) {
  // z = xin @ W                           [Nn, 384]
  gemm(xin, W, nullptr, ws.z, Nn, Kin, CC, s);
  // el / er
  {
    int waves = Nn * HH;
    hipLaunchKernelGGL(attn_lr, dim3((waves + 7) / 8), dim3(256), 0, s,
                       ws.z, al, ar, ws.el, ws.er, Nn);
  }
  // m = -inf, den = 0, x = bias (bias folded into aggregation buffer)
  {
    int thr = Nn * CC;
    hipLaunchKernelGGL(init_agg, dim3((thr + 255) / 256), dim3(256), 0, s,
                       ws.m, ws.den, ws.x, b, Nn);
  }
  hipLaunchKernelGGL(edge_max, dim3((Et + 255) / 256), dim3(256), 0, s,
                     src, dst, ws.el, ws.er, ws.e, ws.m, Et);
  hipLaunchKernelGGL(edge_exp, dim3((Et + 255) / 256), dim3(256), 0, s,
                     dst, ws.e, ws.m, ws.a, ws.den, Et);
  hipLaunchKernelGGL(edge_aggr, dim3((Et + 7) / 8), dim3(256), 0, s,
                     src, dst, ws.a, ws.den, ws.z, ws.x, Et);
}

extern "C" void kernel_launch(void* const* d_in, const int* in_sizes, int n_in,
                              void* d_out, int out_size, void* d_ws, size_t ws_size,
                              hipStream_t stream) {
  const int Nn = in_sizes[4];       // 20000 (gidA length)
  const int Ff = in_sizes[0] / Nn;  // 128
  const int Et = in_sizes[2];       // E + N (self loops included)
  const int Gn = out_size;          // 128 graphs

  const float* feats[2] = { (const float*)d_in[0], (const float*)d_in[1] };
  const int* src[2] = { (const int*)d_in[2], (const int*)d_in[5] };
  const int* dst[2] = { (const int*)d_in[3], (const int*)d_in[6] };
  const int* gid[2] = { (const int*)d_in[4], (const int*)d_in[7] };
  // per-branch params: W1, al1, ar1, b1, Wn, aln, arn, bn, Wl, bl
  const float* P[2][10];
  for (int br = 0; br < 2; ++br)
    for (int i = 0; i < 10; ++i)
      P[br][i] = (const float*)d_in[8 + br * 10 + i];
  const float* Wo = (const float*)d_in[28];
  const float* bo = (const float*)d_in[29];

  // ---- workspace carve-up (floats, 256B-aligned chunks) ----
  float* base = (float*)d_ws;
  size_t off = 0;
  auto carve = [&](size_t nf) -> float* {
    float* p = base + off;
    off += (nf + 63) & ~(size_t)63;
    return p;
  };
  Scratch ws;
  ws.z   = carve((size_t)Nn * CC);
  ws.x   = carve((size_t)Nn * CC);
  float* y = carve((size_t)Nn * DD);
  ws.el  = carve((size_t)Nn * HH);
  ws.er  = carve((size_t)Nn * HH);
  ws.m   = carve((size_t)Nn * HH);
  ws.den = carve((size_t)Nn * HH);
  ws.e   = carve((size_t)Et * HH);
  ws.a   = carve((size_t)Et * HH);
  float* pool[2] = { carve((size_t)Gn * DD), carve((size_t)Gn * DD) };
  float* cnt[2]  = { carve((size_t)Gn),       carve((size_t)Gn)      };
  (void)ws_size;

  for (int br = 0; br < 2; ++br) {
    const float* W1 = P[br][0]; const float* al1 = P[br][1];
    const float* ar1 = P[br][2]; const float* b1 = P[br][3];
    const float* Wn = P[br][4]; const float* aln = P[br][5];
    const float* arn = P[br][6]; const float* bn = P[br][7];
    const float* Wl = P[br][8]; const float* bl = P[br][9];

    // layer 1: gat1 (K = F) then linear1
    run_gat_layer(feats[br], Ff, src[br], dst[br], Et, Nn, W1, al1, ar1, b1, ws, stream);
    gemm(ws.x, Wl, bl, y, Nn, CC, DD, stream);   // y = x @ Wl + bl   [Nn,128]

    // two applications of gatn + linear1 (same weights)
    for (int it = 0; it < 2; ++it) {
      run_gat_layer(y, DD, src[br], dst[br], Et, Nn, Wn, aln, arn, bn, ws, stream);
      gemm(ws.x, Wl, bl, y, Nn, CC, DD, stream);
    }

    // average pooling by graph id
    hipLaunchKernelGGL(zero_buf, dim3((Gn * DD + 255) / 256), dim3(256), 0, stream,
                       pool[br], Gn * DD);
    hipLaunchKernelGGL(zero_buf, dim3((Gn + 255) / 256), dim3(256), 0, stream,
                       cnt[br], Gn);
    hipLaunchKernelGGL(pool_sum, dim3((Nn * DD + 255) / 256), dim3(256), 0, stream,
                       y, gid[br], pool[br], cnt[br], Nn);
  }

  // cat -> linear(256,1) -> float64
  {
    int waves = Gn;
    hipLaunchKernelGGL(final_out, dim3((waves + 7) / 8), dim3(256), 0, stream,
                       pool[0], cnt[0], pool[1], cnt[1], Wo, bo, (double*)d_out, Gn);
  }
}